// BranchingGNN_82403242541640
// MI455X (gfx1250) — compile-verified
//
#include <hip/hip_runtime.h>
#include <hip/hip_bf16.h>
#include <math.h>

typedef __attribute__((ext_vector_type(2))) float v2f;
typedef __attribute__((ext_vector_type(8))) float v8f;

#define H 64
#define LDS_STRIDE 65          // pad 64->65 floats: conflict-free column walks
#define NODES_PER_BLOCK 128    // 8 waves x 16-row tiles
#define THREADS 256

// ---------------------------------------------------------------- utilities

__global__ void zero_f32_kernel(float* __restrict__ p, long long n) {
  long long i = (long long)blockIdx.x * blockDim.x + threadIdx.x;
  long long stride = (long long)gridDim.x * blockDim.x;
  for (; i < n; i += stride) p[i] = 0.f;
}

__global__ void degree_kernel(const int* __restrict__ ev, const int* __restrict__ ec,
                              float* __restrict__ dv, float* __restrict__ dc, int ne) {
  int e = blockIdx.x * blockDim.x + threadIdx.x;
  if (e < ne) {
    atomicAdd(&dv[ev[e]], 1.f);
    atomicAdd(&dc[ec[e]], 1.f);
  }
}

// scatter raw 64-float rows: agg[edst[e]] += hsrc[esrc[e]]  (linearity refactor)
__global__ void scatter_kernel(const float* __restrict__ hsrc,
                               const int* __restrict__ esrc, const int* __restrict__ edst,
                               float* __restrict__ agg, int ne) {
  long long gid = (long long)blockIdx.x * blockDim.x + threadIdx.x;
  long long total = (long long)ne * 16;  // 16 float4 groups per edge
  if (gid >= total) return;
  int e = (int)(gid >> 4);
  int g = (int)(gid & 15);
  int s = esrc[e], d = edst[e];
  const float4* src4 = (const float4*)(hsrc + (long long)s * H);
  float4 v = src4[g];
  float* dst = agg + (long long)d * H + (long long)g * 4;
  atomicAdd(dst + 0, v.x);
  atomicAdd(dst + 1, v.y);
  atomicAdd(dst + 2, v.z);
  atomicAdd(dst + 3, v.w);
}

// --------------------------------------------------- fp32 WMMA 16x16x4 core
// Computes a 16x64 tile of  A(16x64) @ W^T(64x64)  with 16 K-steps of 4.
// ldsA: this wave's 16-row tile, row stride LDS_STRIDE.
// ldsW: W staged as ldsW[n*LDS_STRIDE + k]  (B[k][n] = W[n][k]).
__device__ __forceinline__ void wmma_tile_64(const float* __restrict__ ldsA,
                                             const float* __restrict__ ldsW,
                                             v8f acc[4]) {
  const int lane = threadIdx.x & 31;
  const int m = lane & 15;            // A: row / B: col within 16-tile
  const int kh = (lane >> 4) << 1;    // upper half-wave carries K+2,K+3
#pragma unroll
  for (int k = 0; k < H; k += 4) {
    v2f a;
    a.x = ldsA[m * LDS_STRIDE + k + kh];
    a.y = ldsA[m * LDS_STRIDE + k + kh + 1];
#pragma unroll
    for (int nt = 0; nt < 4; ++nt) {
      const int n = nt * 16 + m;
      v2f b;
      b.x = ldsW[n * LDS_STRIDE + k + kh];
      b.y = ldsW[n * LDS_STRIDE + k + kh + 1];
      // 8 args: (neg_a, A, neg_b, B, c_mod, C, reuse_a, reuse_b)
      acc[nt] = __builtin_amdgcn_wmma_f32_16x16x4_f32(
          false, a, false, b, (short)0, acc[nt], false, false);
    }
  }
}

// ------------------------------------------------------------- encoder
// h = tanh(x @ W1^T + b1) @ W2^T + b2
__global__ void encode_kernel(const float* __restrict__ x, int in_f,
                              const float* __restrict__ W1, const float* __restrict__ b1,
                              const float* __restrict__ W2, const float* __restrict__ b2,
                              float* __restrict__ hout, int n) {
  __shared__ float ldsA[NODES_PER_BLOCK * LDS_STRIDE];
  __shared__ float ldsW[H * LDS_STRIDE];
  const int tid = threadIdx.x;
  const int blockBase = blockIdx.x * NODES_PER_BLOCK;

  for (int i = tid; i < H * H; i += THREADS)
    ldsW[(i >> 6) * LDS_STRIDE + (i & 63)] = W2[i];

  for (int i = tid; i < NODES_PER_BLOCK * H; i += THREADS) {
    int nl = i >> 6, f = i & 63;
    int node = blockBase + nl;
    float acc = 0.f;
    if (node < n) {
      acc = b1[f];
      for (int k = 0; k < in_f; ++k)
        acc += x[(long long)node * in_f + k] * W1[f * in_f + k];
      acc = tanhf(acc);
    }
    ldsA[nl * LDS_STRIDE + f] = acc;
  }
  __syncthreads();

  const int wave = tid >> 5, lane = tid & 31;
  v8f acc[4] = {};
  wmma_tile_64(&ldsA[wave * 16 * LDS_STRIDE], ldsW, acc);

  const int m = lane & 15;
  const int rbase = (lane >> 4) << 3;   // C/D: VGPR j -> row j or j+8
#pragma unroll
  for (int nt = 0; nt < 4; ++nt) {
    const int col = nt * 16 + m;
#pragma unroll
    for (int j = 0; j < 8; ++j) {
      int node = blockBase + wave * 16 + rbase + j;
      if (node < n)
        hout[(long long)node * H + col] = acc[nt][j] + b2[col];
    }
  }
}

// ------------------------------------------------------------- round update
// h = tanh(h + agg @ W^T + deg * b)
__global__ void update_kernel(float* __restrict__ h, const float* __restrict__ agg,
                              const float* __restrict__ W, const float* __restrict__ b,
                              const float* __restrict__ deg, int n) {
  __shared__ float ldsA[NODES_PER_BLOCK * LDS_STRIDE];
  __shared__ float ldsW[H * LDS_STRIDE];
  const int tid = threadIdx.x;
  const int blockBase = blockIdx.x * NODES_PER_BLOCK;

  for (int i = tid; i < H * H; i += THREADS)
    ldsW[(i >> 6) * LDS_STRIDE + (i & 63)] = W[i];

  for (int i = tid; i < NODES_PER_BLOCK * H; i += THREADS) {
    int nl = i >> 6, f = i & 63;
    int node = blockBase + nl;
    ldsA[nl * LDS_STRIDE + f] = (node < n) ? agg[(long long)node * H + f] : 0.f;
  }
  __syncthreads();

  const int wave = tid >> 5, lane = tid & 31;
  v8f acc[4] = {};
  wmma_tile_64(&ldsA[wave * 16 * LDS_STRIDE], ldsW, acc);

  const int m = lane & 15;
  const int rbase = (lane >> 4) << 3;
#pragma unroll
  for (int nt = 0; nt < 4; ++nt) {
    const int col = nt * 16 + m;
#pragma unroll
    for (int j = 0; j < 8; ++j) {
      int node = blockBase + wave * 16 + rbase + j;
      if (node < n) {
        long long idx = (long long)node * H + col;
        h[idx] = tanhf(h[idx] + acc[nt][j] + deg[node] * b[col]);
      }
    }
  }
}

// ------------------------------------------------------------- readout
__global__ void readout_kernel(const float* __restrict__ h, const float* __restrict__ wro,
                               const float* __restrict__ bro, float* __restrict__ out, int n) {
  int v = blockIdx.x * blockDim.x + threadIdx.x;
  if (v >= n) return;
  const float4* h4 = (const float4*)(h + (long long)v * H);
  const float4* w4 = (const float4*)wro;
  float s = bro[0];
#pragma unroll
  for (int g = 0; g < 16; ++g) {
    float4 a = h4[g], w = w4[g];
    s += a.x * w.x + a.y * w.y + a.z * w.z + a.w * w.w;
  }
  out[v] = s;
}

// ------------------------------------------------------------- launch

extern "C" void kernel_launch(void* const* d_in, const int* in_sizes, int n_in,
                              void* d_out, int out_size, void* d_ws, size_t ws_size,
                              hipStream_t stream) {
  const float* var_x   = (const float*)d_in[0];
  const float* con_x   = (const float*)d_in[1];
  const int*   edge_var = (const int*)d_in[2];
  const int*   edge_con = (const int*)d_in[3];
  const float* W_ve1 = (const float*)d_in[4];
  const float* b_ve1 = (const float*)d_in[5];
  const float* W_ve2 = (const float*)d_in[6];
  const float* b_ve2 = (const float*)d_in[7];
  const float* W_ce1 = (const float*)d_in[8];
  const float* b_ce1 = (const float*)d_in[9];
  const float* W_ce2 = (const float*)d_in[10];
  const float* b_ce2 = (const float*)d_in[11];
  const float* W_v2c = (const float*)d_in[12];
  const float* b_v2c = (const float*)d_in[13];
  const float* W_c2v = (const float*)d_in[14];
  const float* b_c2v = (const float*)d_in[15];
  const float* W_ro  = (const float*)d_in[16];
  const float* b_ro  = (const float*)d_in[17];

  const int VF = 7, CF = 5;
  const int n_vars  = in_sizes[0] / VF;
  const int n_cons  = in_sizes[1] / CF;
  const int n_edges = in_sizes[2];
  const int R = in_sizes[12] / (H * H);

  // workspace layout (floats)
  float* h_var = (float*)d_ws;                                // n_vars*H
  float* h_con = h_var + (long long)n_vars * H;               // n_cons*H
  float* agg   = h_con + (long long)n_cons * H;               // n_vars*H (max side)
  float* deg_v = agg + (long long)n_vars * H;                 // n_vars
  float* deg_c = deg_v + n_vars;                              // n_cons (contiguous)

  // degrees (needed for deg*b term of the linearity refactor)
  zero_f32_kernel<<<1024, THREADS, 0, stream>>>(deg_v, (long long)(n_vars + n_cons));
  degree_kernel<<<(n_edges + THREADS - 1) / THREADS, THREADS, 0, stream>>>(
      edge_var, edge_con, deg_v, deg_c, n_edges);

  // encoders (WMMA for the HxH second layer)
  encode_kernel<<<(n_vars + NODES_PER_BLOCK - 1) / NODES_PER_BLOCK, THREADS, 0, stream>>>(
      var_x, VF, W_ve1, b_ve1, W_ve2, b_ve2, h_var, n_vars);
  encode_kernel<<<(n_cons + NODES_PER_BLOCK - 1) / NODES_PER_BLOCK, THREADS, 0, stream>>>(
      con_x, CF, W_ce1, b_ce1, W_ce2, b_ce2, h_con, n_cons);

  long long scatter_threads = (long long)n_edges * 16;
  int scatter_blocks = (int)((scatter_threads + THREADS - 1) / THREADS);

  for (int r = 0; r < R; ++r) {
    // var -> con
    zero_f32_kernel<<<2048, THREADS, 0, stream>>>(agg, (long long)n_cons * H);
    scatter_kernel<<<scatter_blocks, THREADS, 0, stream>>>(h_var, edge_var, edge_con, agg, n_edges);
    update_kernel<<<(n_cons + NODES_PER_BLOCK - 1) / NODES_PER_BLOCK, THREADS, 0, stream>>>(
        h_con, agg, W_v2c + (long long)r * H * H, b_v2c + (long long)r * H, deg_c, n_cons);
    // con -> var
    zero_f32_kernel<<<2048, THREADS, 0, stream>>>(agg, (long long)n_vars * H);
    scatter_kernel<<<scatter_blocks, THREADS, 0, stream>>>(h_con, edge_con, edge_var, agg, n_edges);
    update_kernel<<<(n_vars + NODES_PER_BLOCK - 1) / NODES_PER_BLOCK, THREADS, 0, stream>>>(
        h_var, agg, W_c2v + (long long)r * H * H, b_c2v + (long long)r * H, deg_v, n_vars);
  }

  readout_kernel<<<(n_vars + THREADS - 1) / THREADS, THREADS, 0, stream>>>(
      h_var, W_ro, b_ro, (float*)d_out, n_vars);
}